// digitcap_41034117546513
// MI455X (gfx1250) — compile-verified
//
#include <hip/hip_runtime.h>
#include <math.h>

// ---------------------------------------------------------------------------
// CapsNet dynamic routing, fused (x_hat never materialized).
//   x      : (256, 1152, 8)  f32
//   weight : (10, 1152, 16, 8) f32
//   out    : (256, 10, 16)   f32
// Core math s[b,o,:] = sum_i c[b,o,i] * (x[b,i,:] @ W[o,i,:,:]^T) runs on
// V_WMMA_F32_16X16X4_F32: per (16-batch tile, o), a chain of (16x8)@(8x16)
// GEMMs (two K=4 chunks per capsule) accumulated in a 16x16 f32 C tile.
// Total footprint (x 9.4MB + W 5.9MB + b 11.8MB) is L2-resident on MI455X;
// recomputing x_hat avoids the 189MB intermediate entirely.
// ---------------------------------------------------------------------------

typedef float v2f __attribute__((ext_vector_type(2)));
typedef float v8f __attribute__((ext_vector_type(8)));

#define BATCH 256
#define ICAPS 1152
#define OCAPS 10
#define DIN   8
#define DOUT  16
#define ISPLIT 4
#define I_PER_WAVE (ICAPS / (ISPLIT * 8))   // 36 capsules per wave

__global__ __launch_bounds__(256) void zero_f32(float* __restrict__ p, int n) {
  int i = blockIdx.x * 256 + threadIdx.x;
  if (i < n) p[i] = 0.0f;
}

// s[b,o,d] += sum_i c[b,o,i] * sum_k W[o,i,d,k] * x[b,i,k]   via WMMA f32 16x16x4.
// Grid: (B/16, O, ISPLIT), block 256 (8 waves). Each wave owns 36 capsules,
// accumulates a 16x16 (batch x dout) f32 tile in registers, then atomically
// adds its partial into the 160KB s buffer (L2-resident).
// MODE is a compile-time constant so the c-coefficient path has no branches
// in the WMMA feed loop.
template <int MODE>
__global__ __launch_bounds__(256) void s_wmma_kernel(
    const float* __restrict__ x, const float* __restrict__ w,
    const float* __restrict__ blog, const float* __restrict__ mx,
    const float* __restrict__ rse, float* __restrict__ s)
{
  const int btile = blockIdx.x;            // 0..15
  const int o     = blockIdx.y;            // 0..9
  const int wave  = threadIdx.x >> 5;
  const int lane  = threadIdx.x & 31;
  const int row   = lane & 15;             // A: M index fed by this lane; B: N index
  const int hi    = lane >> 4;             // k-pair selector within a K=4 chunk
  const int b     = btile * 16 + row;
  const int i0    = (blockIdx.z * 8 + wave) * I_PER_WAVE;

  v8f acc = {0.f, 0.f, 0.f, 0.f, 0.f, 0.f, 0.f, 0.f};

  // Hoisted per-lane base pointers; advance by constant strides so the hot
  // loop uses immediate-offset global_load_b64 with no 64-bit address math.
  const float* xr = x + ((size_t)b * ICAPS + i0) * DIN + hi * 2;          // x[b,i,2hi..]
  const float* wr = w + ((size_t)(o * ICAPS + i0) * DOUT + row) * DIN + hi * 2; // W[o,i,row,2hi..]
  const float* bp = (MODE ? blog + ((size_t)b * OCAPS + o) * ICAPS + i0 : nullptr);
  const float* mp = (MODE ? mx  + (size_t)o * ICAPS + i0 : nullptr);
  const float* rp = (MODE ? rse + (size_t)o * ICAPS + i0 : nullptr);

  #pragma unroll 4
  for (int ii = 0; ii < I_PER_WAVE; ++ii) {
    // routing coefficient c[b,o,i] (softmax over the BATCH axis — torch legacy dim)
    float c;
    if (MODE) {
      c = __expf(bp[ii] - mp[ii]) * rp[ii];
    } else {
      c = 1.0f / 256.0f;                   // iteration 0: b == 0 everywhere
    }

    v2f a0 = *(const v2f*)(xr);            // K-chunk 0 (k = 0..3)
    v2f b0 = *(const v2f*)(wr);
    v2f a1 = *(const v2f*)(xr + 4);        // K-chunk 1 (k = 4..7)
    v2f b1 = *(const v2f*)(wr + 4);
    a0 *= c;
    a1 *= c;
    acc = __builtin_amdgcn_wmma_f32_16x16x4_f32(false, a0, false, b0,
                                                (short)0, acc, false, false);
    acc = __builtin_amdgcn_wmma_f32_16x16x4_f32(false, a1, false, b1,
                                                (short)0, acc, false, false);

    xr += DIN;
    wr += DOUT * DIN;
    __builtin_prefetch(wr, 0, 3);          // next capsule's W row, WGP-scope
  }

  // D layout: lane holds column d = lane%16, rows m = r + 8*hi.
  #pragma unroll
  for (int r = 0; r < 8; ++r) {
    const int bo = btile * 16 + hi * 8 + r;
    atomicAdd(&s[((size_t)bo * OCAPS + o) * DOUT + row], acc[r]);
  }
}

// squash over Dout: scale = n^2/(1+n^2)/(n+1e-8)
__global__ __launch_bounds__(256) void squash_kernel(const float* __restrict__ s,
                                                     float* __restrict__ vout) {
  int idx = blockIdx.x * 256 + threadIdx.x;   // flattened (b, o)
  if (idx >= BATCH * OCAPS) return;
  const float* sp = s + (size_t)idx * DOUT;
  float nn = 0.f;
  #pragma unroll
  for (int d = 0; d < DOUT; ++d) nn = fmaf(sp[d], sp[d], nn);
  const float nrm = sqrtf(nn);
  const float scale = nn / (1.0f + nn) / (nrm + 1e-8f);
  #pragma unroll
  for (int d = 0; d < DOUT; ++d) vout[(size_t)idx * DOUT + d] = scale * sp[d];
}

// b[b,o,i] (+)= sum_d v[b,o,d] * x_hat[b,o,i,d], x_hat recomputed from W (LDS) and x.
// Grid (I, O), block 256 = one thread per batch.
__global__ __launch_bounds__(256) void bupd_kernel(
    const float* __restrict__ x, const float* __restrict__ w,
    const float* __restrict__ v, float* __restrict__ blog, int first)
{
  const int i = blockIdx.x, o = blockIdx.y;
  __shared__ float Ws[DOUT * DIN];
  const int t = threadIdx.x;
  if (t < DOUT * DIN) Ws[t] = w[(size_t)(o * ICAPS + i) * (DOUT * DIN) + t];
  __syncthreads();

  const int b = t;
  float xr[DIN];
  const float* xp = x + ((size_t)b * ICAPS + i) * DIN;
  #pragma unroll
  for (int k = 0; k < DIN; ++k) xr[k] = xp[k];
  const float* vp = v + ((size_t)b * OCAPS + o) * DOUT;

  float acc = 0.f;
  #pragma unroll
  for (int d = 0; d < DOUT; ++d) {
    float xh = 0.f;
    #pragma unroll
    for (int k = 0; k < DIN; ++k) xh = fmaf(Ws[d * DIN + k], xr[k], xh);
    acc = fmaf(xh, vp[d], acc);
  }
  const size_t bi = ((size_t)b * OCAPS + o) * ICAPS + i;
  blog[bi] = first ? acc : (blog[bi] + acc);
}

// Per (o,i): max_b b[b,o,i] and 1/sum_b exp(b - max). Block 256 = one thread per batch.
__global__ __launch_bounds__(256) void stats_kernel(const float* __restrict__ blog,
                                                    float* __restrict__ mx,
                                                    float* __restrict__ rse) {
  const int oi = blockIdx.x;               // 0..11519
  const int o = oi / ICAPS, i = oi - o * ICAPS;
  const int t = threadIdx.x;
  __shared__ float red[256];

  const float bv = blog[((size_t)t * OCAPS + o) * ICAPS + i];
  red[t] = bv;
  __syncthreads();
  for (int s = 128; s > 0; s >>= 1) {
    if (t < s) red[t] = fmaxf(red[t], red[t + s]);
    __syncthreads();
  }
  const float m = red[0];
  __syncthreads();
  red[t] = __expf(bv - m);
  __syncthreads();
  for (int s = 128; s > 0; s >>= 1) {
    if (t < s) red[t] += red[t + s];
    __syncthreads();
  }
  if (t == 0) { mx[oi] = m; rse[oi] = 1.0f / red[0]; }
}

extern "C" void kernel_launch(void* const* d_in, const int* in_sizes, int n_in,
                              void* d_out, int out_size, void* d_ws, size_t ws_size,
                              hipStream_t stream) {
  const float* x = (const float*)d_in[0];
  const float* w = (const float*)d_in[1];
  float* out = (float*)d_out;

  // workspace layout (~12.2 MB total)
  float* b_ws   = (float*)d_ws;                                 // B*O*I routing logits
  float* s_ws   = b_ws  + (size_t)BATCH * OCAPS * ICAPS;        // B*O*Dout
  float* v_ws   = s_ws  + (size_t)BATCH * OCAPS * DOUT;         // B*O*Dout
  float* mx_ws  = v_ws  + (size_t)BATCH * OCAPS * DOUT;         // O*I softmax max
  float* rse_ws = mx_ws + (size_t)OCAPS * ICAPS;                // O*I 1/sumexp

  const dim3 sgrid(BATCH / 16, OCAPS, ISPLIT);
  const dim3 sblk(256);
  const int sN = BATCH * OCAPS * DOUT;
  const dim3 bgrid(ICAPS, OCAPS);

  // ---- routing iteration 0 (uniform c = 1/256) ----
  zero_f32<<<(sN + 255) / 256, 256, 0, stream>>>(s_ws, sN);
  s_wmma_kernel<0><<<sgrid, sblk, 0, stream>>>(x, w, nullptr, nullptr, nullptr, s_ws);
  squash_kernel<<<(BATCH * OCAPS + 255) / 256, 256, 0, stream>>>(s_ws, v_ws);
  bupd_kernel<<<bgrid, 256, 0, stream>>>(x, w, v_ws, b_ws, 1);
  stats_kernel<<<OCAPS * ICAPS, 256, 0, stream>>>(b_ws, mx_ws, rse_ws);

  // ---- routing iteration 1 ----
  zero_f32<<<(sN + 255) / 256, 256, 0, stream>>>(s_ws, sN);
  s_wmma_kernel<1><<<sgrid, sblk, 0, stream>>>(x, w, b_ws, mx_ws, rse_ws, s_ws);
  squash_kernel<<<(BATCH * OCAPS + 255) / 256, 256, 0, stream>>>(s_ws, v_ws);
  bupd_kernel<<<bgrid, 256, 0, stream>>>(x, w, v_ws, b_ws, 0);
  stats_kernel<<<OCAPS * ICAPS, 256, 0, stream>>>(b_ws, mx_ws, rse_ws);

  // ---- routing iteration 2 (final; squash straight to output) ----
  zero_f32<<<(sN + 255) / 256, 256, 0, stream>>>(s_ws, sN);
  s_wmma_kernel<1><<<sgrid, sblk, 0, stream>>>(x, w, b_ws, mx_ws, rse_ws, s_ws);
  squash_kernel<<<(BATCH * OCAPS + 255) / 256, 256, 0, stream>>>(s_ws, out);
}